// GEVBeamformer_55748675502301
// MI455X (gfx1250) — compile-verified
//
#include <hip/hip_runtime.h>

typedef __attribute__((ext_vector_type(2))) float v2f;
typedef __attribute__((ext_vector_type(8))) float v8f;

#define GEV_B 4
#define GEV_M 8
#define GEV_F 513
#define GEV_T 2000
#define GEV_EPS 1e-6f
#define POWER_ITERS 40

// ---------------------------------------------------------------------------
// Kernel 1: per-(b,f) 8x8 generalized eigen solve -> beamformer weights w[8].
// One thread per (b,f). SCM layout is (B, M, M, F): consecutive threads read
// consecutive f => fully coalesced. All loops unrolled => arrays stay in VGPRs.
// ---------------------------------------------------------------------------
__global__ void gev_weights_kernel(const float* __restrict__ tscm,
                                   const float* __restrict__ nscm,
                                   float* __restrict__ wout) {
  const int tid = blockIdx.x * blockDim.x + threadIdx.x;
  const int total = GEV_B * GEV_F;
  if (tid >= total) return;
  const int b = tid / GEV_F;
  const int f = tid - b * GEV_F;

  float a[GEV_M][GEV_M];   // target SCM, later overwritten with cmat rows
  float n[GEV_M][GEV_M];   // noise SCM, later holds Cholesky factor L (lower)

#pragma unroll
  for (int i = 0; i < GEV_M; ++i) {
#pragma unroll
    for (int j = 0; j < GEV_M; ++j) {
      const long idx = (((long)b * GEV_M + i) * GEV_M + j) * GEV_F + f;
      a[i][j] = tscm[idx];
      n[i][j] = nscm[idx];
    }
  }

  // Condition noise SCM: (n + eps*tr(n)/M * I) / (1+eps)
  float tr = 0.f;
#pragma unroll
  for (int i = 0; i < GEV_M; ++i) tr += n[i][i];
  const float diag_add = GEV_EPS * tr / (float)GEV_M;
  const float scl = 1.f / (1.f + GEV_EPS);
#pragma unroll
  for (int i = 0; i < GEV_M; ++i) {
#pragma unroll
    for (int j = 0; j < GEV_M; ++j) {
      n[i][j] = (n[i][j] + ((i == j) ? diag_add : 0.f)) * scl;
    }
  }

  // In-place Cholesky: n -> L (lower triangular)
#pragma unroll
  for (int k = 0; k < GEV_M; ++k) {
    float d = n[k][k];
#pragma unroll
    for (int p = 0; p < GEV_M; ++p) {
      if (p < k) d -= n[k][p] * n[k][p];
    }
    d = sqrtf(fmaxf(d, 1e-20f));
    n[k][k] = d;
    const float inv = 1.f / d;
#pragma unroll
    for (int i = 0; i < GEV_M; ++i) {
      if (i > k) {
        float s = n[i][k];
#pragma unroll
        for (int p = 0; p < GEV_M; ++p) {
          if (p < k) s -= n[i][p] * n[k][p];
        }
        n[i][k] = s * inv;
      }
    }
  }

  // X = Linv * a  (forward substitution, column-wise), stored back into a
#pragma unroll
  for (int j = 0; j < GEV_M; ++j) {
#pragma unroll
    for (int i = 0; i < GEV_M; ++i) {
      float s = a[i][j];
#pragma unroll
      for (int p = 0; p < GEV_M; ++p) {
        if (p < i) s -= n[i][p] * a[p][j];
      }
      a[i][j] = s / n[i][i];
    }
  }

  // cmat = X * Linv^T: cmat[r][:] = Linv * X[r][:]^T  (forward subst per row)
#pragma unroll
  for (int r = 0; r < GEV_M; ++r) {
    float y[GEV_M];
#pragma unroll
    for (int i = 0; i < GEV_M; ++i) {
      float s = a[r][i];
#pragma unroll
      for (int p = 0; p < GEV_M; ++p) {
        if (p < i) s -= n[i][p] * y[p];
      }
      y[i] = s / n[i][i];
    }
#pragma unroll
    for (int i = 0; i < GEV_M; ++i) a[r][i] = y[i];
  }

  // Power iteration on cmat (PSD => dominant eigval is max eigval)
  float v[GEV_M];
#pragma unroll
  for (int i = 0; i < GEV_M; ++i) v[i] = 1.f;
  for (int it = 0; it < POWER_ITERS; ++it) {
    float u[GEV_M];
    float nrm2 = 0.f;
#pragma unroll
    for (int i = 0; i < GEV_M; ++i) {
      float s = 0.f;
#pragma unroll
      for (int k = 0; k < GEV_M; ++k) s += a[i][k] * v[k];
      u[i] = s;
      nrm2 += s * s;
    }
    const float rn = rsqrtf(fmaxf(nrm2, 1e-30f));
#pragma unroll
    for (int i = 0; i < GEV_M; ++i) v[i] = u[i] * rn;
  }

  // w = Linv^T v  <=>  solve L^T w = v  (backward substitution)
  float w[GEV_M];
#pragma unroll
  for (int ii = 0; ii < GEV_M; ++ii) {
    const int i = GEV_M - 1 - ii;
    float s = v[i];
#pragma unroll
    for (int p = 0; p < GEV_M; ++p) {
      if (p > i) s -= n[p][i] * w[p];
    }
    w[i] = s / n[i][i];
  }

  float nrm2 = 0.f;
#pragma unroll
  for (int i = 0; i < GEV_M; ++i) nrm2 += w[i] * w[i];
  const float rn = rsqrtf(fmaxf(nrm2, 1e-30f));
#pragma unroll
  for (int i = 0; i < GEV_M; ++i) wout[tid * GEV_M + i] = w[i] * rn;
}

// ---------------------------------------------------------------------------
// Kernel 2: apply beamformer, out[b,f,t] = sum_m w[b,f,m] * mix[b,m,f,t].
// WMMA mapping per 16-t tile (V_WMMA_F32_16X16X4_F32, two chained for K=8):
//   A (16x4) = weights broadcast across all 16 rows
//     (lanes 0-15 hold K=0,1 in VGPR0/1; lanes 16-31 hold K=2,3)
//   B (4x16) = mix rows, B[k][j] = mix[mb+k, t0+j]
//     (VGPR0: K=0 on lanes 0-15 / K=2 on lanes 16-31; VGPR1: K=1 / K=3)
//   D rows are all identical = out[t0..t0+15]; lanes 0-15 store row M=0.
// 4 waves per block cover interleaved 16-t tiles of one (b,f).
// ---------------------------------------------------------------------------
__global__ void __launch_bounds__(128)
gev_apply_kernel(const float* __restrict__ mix,
                 const float* __restrict__ wv,
                 float* __restrict__ out) {
  const int lane = threadIdx.x & 31;
  const int wid = threadIdx.x >> 5;          // 0..3
  const int bf = blockIdx.x;                 // 0 .. B*F-1
  const int b = bf / GEV_F;
  const int f = bf - b * GEV_F;

  // Broadcast weights (uniform across wave -> scalar loads)
  const float* wp = wv + (long)bf * GEV_M;
  const float w0 = wp[0], w1 = wp[1], w2 = wp[2], w3 = wp[3];
  const float w4 = wp[4], w5 = wp[5], w6 = wp[6], w7 = wp[7];

  const bool hi = lane >= 16;
  v2f A0, A1;
  A0.x = hi ? w2 : w0;  A0.y = hi ? w3 : w1;   // K-block 0..3
  A1.x = hi ? w6 : w4;  A1.y = hi ? w7 : w5;   // K-block 4..7

  const long mstride = (long)GEV_F * GEV_T;
  const float* base = mix + (long)b * GEV_M * mstride + (long)f * GEV_T;
  const int r = hi ? 2 : 0;
  const float* p0 = base + (long)(r + 0) * mstride;  // B0.x source row
  const float* p1 = base + (long)(r + 1) * mstride;  // B0.y source row
  const float* p2 = base + (long)(r + 4) * mstride;  // B1.x source row
  const float* p3 = base + (long)(r + 5) * mstride;  // B1.y source row

  float* op = out + (long)bf * GEV_T;
  const int tcol = lane & 15;
  const int ntiles = GEV_T / 16;             // 2000/16 = 125, no tail

  for (int tile = wid; tile < ntiles; tile += 4) {
    const int t = tile * 16 + tcol;

    // Prefetch this wave's next tile (t + 64) into the cache hierarchy
    __builtin_prefetch(p0 + t + 64, 0, 0);
    __builtin_prefetch(p1 + t + 64, 0, 0);
    __builtin_prefetch(p2 + t + 64, 0, 0);
    __builtin_prefetch(p3 + t + 64, 0, 0);

    v2f B0, B1;
    B0.x = p0[t];  B0.y = p1[t];
    B1.x = p2[t];  B1.y = p3[t];

    v8f c = {};
    c = __builtin_amdgcn_wmma_f32_16x16x4_f32(false, A0, false, B0,
                                              (short)0, c, false, false);
    c = __builtin_amdgcn_wmma_f32_16x16x4_f32(false, A1, false, B1,
                                              (short)0, c, false, false);

    if (lane < 16) {
      op[t] = c[0];                          // row M=0, column N=lane
    }
  }
}

// ---------------------------------------------------------------------------
extern "C" void kernel_launch(void* const* d_in, const int* in_sizes, int n_in,
                              void* d_out, int out_size, void* d_ws, size_t ws_size,
                              hipStream_t stream) {
  const float* mix = (const float*)d_in[0];        // (B, M, F, T) f32
  const float* tscm = (const float*)d_in[1];       // (B, M, M, F) f32
  const float* nscm = (const float*)d_in[2];       // (B, M, M, F) f32
  float* out = (float*)d_out;                      // (B, F, T) f32
  float* wbuf = (float*)d_ws;                      // B*F*8 floats scratch

  const int total_bf = GEV_B * GEV_F;              // 2052

  // Phase 1: beamformer weights
  {
    const int threads = 128;
    const int blocks = (total_bf + threads - 1) / threads;
    gev_weights_kernel<<<blocks, threads, 0, stream>>>(tscm, nscm, wbuf);
  }

  // Phase 2: WMMA streaming apply (bandwidth-bound: ~148 MB total traffic)
  {
    gev_apply_kernel<<<total_bf, 128, 0, stream>>>(mix, wbuf, out);
  }
}